// PPM_14757507629246
// MI455X (gfx1250) — compile-verified
//
#include <hip/hip_runtime.h>

typedef __attribute__((ext_vector_type(2))) float v2f;
typedef __attribute__((ext_vector_type(8))) float v8f;

#define N_ELEMS  5
#define N_LINESK 12
#define N_CHUNK  32
#define FL_TOTAL (N_LINESK * 8192)
#define LSTRIDE  518   // padded LDS row stride (floats): even (8B-aligned b64), 518%64=6 -> conflict-free

// Bilinear rotation sampling, exactly mirroring the reference formulas (f32).
__device__ __forceinline__ void bilerp_setup(float yf, float xf, float cth, float sth,
                                             int& i00, int& i01, int& i10, int& i11,
                                             float& w00, float& w01, float& w10, float& w11) {
  const float invW = 1.0f / 512.0f;
  float gx = (2.0f * xf + 1.0f) * invW - 1.0f;
  float gy = (2.0f * yf + 1.0f) * invW - 1.0f;
  float x_in = cth * gx - sth * gy;
  float y_in = sth * gx + cth * gy;
  float ix = fminf(fmaxf(((x_in + 1.0f) * 512.0f - 1.0f) * 0.5f, 0.0f), 511.0f);
  float iy = fminf(fmaxf(((y_in + 1.0f) * 512.0f - 1.0f) * 0.5f, 0.0f), 511.0f);
  float x0f = floorf(ix), y0f = floorf(iy);
  float wx = ix - x0f, wy = iy - y0f;
  int x0 = (int)x0f, y0 = (int)y0f;
  int x1 = min(x0 + 1, 511), y1 = min(y0 + 1, 511);
  i00 = y0 * 512 + x0; i01 = y0 * 512 + x1;
  i10 = y1 * 512 + x0; i11 = y1 * 512 + x1;
  w00 = (1.0f - wx) * (1.0f - wy);
  w01 = wx * (1.0f - wy);
  w10 = (1.0f - wx) * wy;
  w11 = wx * wy;
}

__global__ __launch_bounds__(256) void ppm_fused(
    const float* __restrict__ xp, const float* __restrict__ wel,
    const float* __restrict__ dfu, const float* __restrict__ thetap,
    const int* __restrict__ lidx, float* __restrict__ out)
{
  __shared__ float lacatt[16 * LSTRIDE];     // lac in phases 1-3, attenuation afterwards
  __shared__ float chunktot[N_CHUNK * 16];
  __shared__ float carry[N_CHUNK * 16];
  __shared__ float sred[N_ELEMS * 16];

  const int tid  = threadIdx.x;
  const int lane = tid & 31;
  const int wave = tid >> 5;
  const int t    = blockIdx.x;          // tile: 16 rows b = [16t, 16t+16)
  const int c    = t >> 5;              // image row-block (N_ROWS dim)
  const int ybase = (t & 31) << 4;      // y within 512x512 image
  const int bbase = t << 4;

  const float theta = thetap[0];
  const float cth = cosf(theta), sth = sinf(theta);
  float w[N_ELEMS];
  #pragma unroll
  for (int e = 0; e < N_ELEMS; ++e) w[e] = wel[e];

  const float SCALE = 0.01f / 512.0f;

  // ---- Phase 1: lac -> LDS, per-row chunk totals -> LDS --------------------
  {
    const int row = tid & 15;
    const int cp  = tid >> 4;                 // 0..15, two chunks each
    const float yf = (float)(ybase + row);
    #pragma unroll
    for (int h = 0; h < 2; ++h) {
      const int chunk = cp * 2 + h;
      float tot = 0.0f;
      #pragma unroll 4
      for (int k = 0; k < 16; ++k) {
        const int x = (chunk << 4) + k;
        int i00, i01, i10, i11; float w00, w01, w10, w11;
        bilerp_setup(yf, (float)x, cth, sth, i00, i01, i10, i11, w00, w01, w10, w11);
        float lac = 0.0f;
        #pragma unroll
        for (int e = 0; e < N_ELEMS; ++e) {
          const float* img = xp + ((size_t)(e * 16 + c) << 18);
          float conc = img[i00] * w00 + img[i01] * w01 + img[i10] * w10 + img[i11] * w11;
          lac = fmaf(w[e], conc, lac);
        }
        lacatt[row * LSTRIDE + x] = lac;
        tot += lac;
      }
      chunktot[chunk * 16 + row] = tot;
    }
  }
  __syncthreads();

  // ---- Phase 2: exclusive scan of chunk totals -> carries; transmission ----
  if (tid < 16) {
    float acc = 0.0f;
    #pragma unroll 1
    for (int ch = 0; ch < N_CHUNK; ++ch) {
      carry[ch * 16 + tid] = acc;
      acc += chunktot[ch * 16 + tid];
    }
    out[FL_TOTAL + bbase + tid] = acc * SCALE;   // transmission[b]
  }
  __syncthreads();

  // ---- Phase 3: WMMA f32 scan per chunk; attenuation overwrites lac in LDS -
  {
    const int n      = lane & 15;             // B/C/D column
    const int hi2    = (lane >> 4) << 1;      // A/B K striping: +0 or +2
    const int rowA   = lane & 15;             // A-layout row
    const int rowOff = (lane >> 4) << 3;      // C/D rows: r (+8 upper lanes)

    #pragma unroll
    for (int ci = 0; ci < 4; ++ci) {
      const int chunk = (wave << 2) + ci;     // each wave owns 4 chunks
      float cc[8];
      #pragma unroll
      for (int r = 0; r < 8; ++r) cc[r] = carry[chunk * 16 + r + rowOff];
      v8f d;
      #pragma unroll
      for (int r = 0; r < 8; ++r) d[r] = cc[r];
      // D = carry + A * U   (U[k][n] = 1 iff k <= n)
      #pragma unroll
      for (int p = 0; p < 4; ++p) {
        const float* ap = &lacatt[rowA * LSTRIDE + (chunk << 4) + 4 * p + hi2];
        v2f a, b;
        a.x = ap[0]; a.y = ap[1];
        const int k0 = 4 * p + hi2;
        b.x = (k0     <= n) ? 1.0f : 0.0f;
        b.y = (k0 + 1 <= n) ? 1.0f : 0.0f;
        d = __builtin_amdgcn_wmma_f32_16x16x4_f32(false, a, false, b, (short)0, d, false, false);
      }
      // exclusive prefix = inclusive shifted one column (col 0 <- carry); atten -> LDS
      #pragma unroll
      for (int r = 0; r < 8; ++r) {
        float prev = __int_as_float(
            __builtin_amdgcn_ds_bpermute((lane - 1) << 2, __float_as_int(d[r])));
        float excl = (n == 0) ? cc[r] : prev;
        lacatt[(r + rowOff) * LSTRIDE + (chunk << 4) + n] = __expf(-SCALE * excl);
      }
    }
  }
  __syncthreads();

  // ---- Phase 4: S_e[row] = sum_s atten * conc  (conc regathered; L2-resident)
  {
    const int row = tid >> 4;                 // 16 threads per row
    const int i16 = tid & 15;                 // column-strip id (== lane&15)
    const float yf = (float)(ybase + row);
    float acc[N_ELEMS];
    #pragma unroll
    for (int e = 0; e < N_ELEMS; ++e) acc[e] = 0.0f;

    #pragma unroll 1
    for (int jj = 0; jj < 32; ++jj) {
      const int x = i16 + (jj << 4);          // stride-16: conflict-free LDS, coalesced gathers
      const float att = lacatt[row * LSTRIDE + x];
      int i00, i01, i10, i11; float w00, w01, w10, w11;
      bilerp_setup(yf, (float)x, cth, sth, i00, i01, i10, i11, w00, w01, w10, w11);
      #pragma unroll
      for (int e = 0; e < N_ELEMS; ++e) {
        const float* img = xp + ((size_t)(e * 16 + c) << 18);
        float conc = img[i00] * w00 + img[i01] * w01 + img[i10] * w10 + img[i11] * w11;
        acc[e] = fmaf(att, conc, acc[e]);
      }
    }
    // butterfly over the 16 threads of this row (within a 16-lane half-wave)
    #pragma unroll
    for (int e = 0; e < N_ELEMS; ++e) {
      float v = acc[e];
      v += __int_as_float(__builtin_amdgcn_ds_swizzle(__float_as_int(v), 0x041F)); // xor 1
      v += __int_as_float(__builtin_amdgcn_ds_swizzle(__float_as_int(v), 0x081F)); // xor 2
      v += __int_as_float(__builtin_amdgcn_ds_swizzle(__float_as_int(v), 0x101F)); // xor 4
      v += __int_as_float(__builtin_amdgcn_ds_swizzle(__float_as_int(v), 0x201F)); // xor 8
      if (i16 == 0) sred[e * 16 + row] = v;
    }
  }
  __syncthreads();

  // ---- Phase 5: fl_sig[l][b] = 1e5 * 0.0017 * dfu[l] * S_{e_l}[b] ----------
  if (tid < N_LINESK * 16) {
    const int l = tid >> 4, r = tid & 15;
    const int e = lidx[l];
    out[l * 8192 + bbase + r] = (100000.0f * 0.0017f) * dfu[l] * sred[e * 16 + r];
  }
}

extern "C" void kernel_launch(void* const* d_in, const int* in_sizes, int n_in,
                              void* d_out, int out_size, void* d_ws, size_t ws_size,
                              hipStream_t stream) {
  (void)in_sizes; (void)n_in; (void)d_ws; (void)ws_size; (void)out_size;
  const float* xp    = (const float*)d_in[0];
  const float* wel   = (const float*)d_in[1];
  const float* dfu   = (const float*)d_in[2];
  const float* theta = (const float*)d_in[3];
  const int*   lidx  = (const int*)d_in[4];
  float* out = (float*)d_out;
  ppm_fused<<<512, 256, 0, stream>>>(xp, wel, dfu, theta, lidx, out);
}